// GCN_37924561224136
// MI455X (gfx1250) — compile-verified
//
#include <hip/hip_runtime.h>
#include <hip/hip_bf16.h>

typedef float v2f __attribute__((ext_vector_type(2)));
typedef float v8f __attribute__((ext_vector_type(8)));

#define N_NODES 200000
#define N_EDGES 1200000
#define F_IN 10
#define HID 128
#define N_CLS 10

// ---------------------------------------------------------------------------
// Kernel 0: out[n, c] = b2[c]   (atomic accumulation target must be init'd)
// ---------------------------------------------------------------------------
__global__ void gcn_init_out(const float* __restrict__ b2, float* __restrict__ out, int n) {
    int i = blockIdx.x * blockDim.x + threadIdx.x;
    if (i < n) {
        int c = i - (i / N_CLS) * N_CLS;
        out[i] = b2[c];
    }
}

// ---------------------------------------------------------------------------
// Kernel 1: h = relu(x @ W1 + b1), fp32 WMMA 16x16x4.
// One wave per 16 nodes. K=10 padded to 12 (3 k-steps of 4). 8 N-tiles of 16.
// A layout (16x4 f32): lane<16 -> {K0,K1}, lane>=16 -> {K2,K3}, M = lane&15.
// B layout (4x16 f32): col = lane&15; vgpr v holds K = kstep*4 + (lane>>4)*2 + v.
// C/D layout: vgpr r -> M = r + (lane>>4)*8, N = lane&15.
// ---------------------------------------------------------------------------
__global__ void gcn_lin1_relu(const float* __restrict__ x,
                              const float* __restrict__ W1,
                              const float* __restrict__ b1,
                              float* __restrict__ h) {
    const int lane    = threadIdx.x & 31;
    const int wave    = blockIdx.x * (blockDim.x >> 5) + (threadIdx.x >> 5);
    const int row0    = wave * 16;
    const int halfsel = lane >> 4;
    const int l15     = lane & 15;

    // Load A fragments (branchless zero-padding for K=10,11)
    v2f a[3];
    const int arow = row0 + l15;
#pragma unroll
    for (int kk = 0; kk < 3; ++kk) {
        int k  = kk * 4 + halfsel * 2;
        int k0 = (k     < F_IN) ? k     : 0;
        int k1 = (k + 1 < F_IN) ? k + 1 : 0;
        float m0 = (k     < F_IN) ? 1.0f : 0.0f;
        float m1 = (k + 1 < F_IN) ? 1.0f : 0.0f;
        v2f av;
        av.x = x[arow * F_IN + k0] * m0;
        av.y = x[arow * F_IN + k1] * m1;
        a[kk] = av;
    }

#pragma unroll
    for (int nt = 0; nt < 8; ++nt) {
        const int col = nt * 16 + l15;
        v8f acc = {0.f, 0.f, 0.f, 0.f, 0.f, 0.f, 0.f, 0.f};
#pragma unroll
        for (int kk = 0; kk < 3; ++kk) {
            int k  = kk * 4 + halfsel * 2;
            int k0 = (k     < F_IN) ? k     : 0;
            int k1 = (k + 1 < F_IN) ? k + 1 : 0;
            float m0 = (k     < F_IN) ? 1.0f : 0.0f;
            float m1 = (k + 1 < F_IN) ? 1.0f : 0.0f;
            v2f bv;
            bv.x = W1[k0 * HID + col] * m0;
            bv.y = W1[k1 * HID + col] * m1;
            acc = __builtin_amdgcn_wmma_f32_16x16x4_f32(
                false, a[kk], false, bv, (short)0, acc, false, false);
        }
        const float bias = b1[col];
#pragma unroll
        for (int r = 0; r < 8; ++r) {
            const int m = r + halfsel * 8;
            float v = acc[r] + bias;
            v = v > 0.0f ? v : 0.0f;
            h[(long long)(row0 + m) * HID + col] = v;
        }
    }
}

// ---------------------------------------------------------------------------
// Kernel 2: for each edge e: out[dst[e], :] += h[src[e], :] @ W2
// One wave per 16 edges. A = 16 gathered h rows (16x128 f32), B = W2 padded
// to 128x16, K=128 -> 32 fp32 WMMA k-steps. D rows scattered with hardware
// f32 atomics (12M atomics total vs 153.6M for the naive 128-wide scheme).
// h (102.4 MB) stays resident in the 192 MB L2 across the whole phase.
//
// W2 is staged ONCE per block into LDS, transposed + zero-padded:
//   w2s[c][k] = (c < 10) ? W2[k][c] : 0,  c in [0,16), k in [0,128)
// stride 132 floats so the 16 lanes of a half-wave (addr stride = 4 banks,
// footprint 2 banks per ds_load_b64) are bank-conflict-free. This removes
// ~64 global loads + 64 masking v_muls per wave from the hot kernel.
// ---------------------------------------------------------------------------
#define W2S_STRIDE 132

__global__ void gcn_edge_scatter(const float* __restrict__ h,
                                 const long long* __restrict__ src,
                                 const long long* __restrict__ dst,
                                 const float* __restrict__ W2,
                                 float* __restrict__ out) {
    __shared__ float w2s[16][W2S_STRIDE];

    // Stage padded/transposed W2: 2048 elements, 8 per thread (256 threads).
    for (int i = threadIdx.x; i < 16 * HID; i += blockDim.x) {
        const int c = i >> 7;       // 0..15  (B column)
        const int k = i & 127;      // 0..127 (K index)
        w2s[c][k] = (c < N_CLS) ? W2[k * N_CLS + c] : 0.0f;
    }
    __syncthreads();

    const int lane    = threadIdx.x & 31;
    const int wave    = blockIdx.x * (blockDim.x >> 5) + (threadIdx.x >> 5);
    const long long ebase = (long long)wave * 16;
    const int halfsel = lane >> 4;
    const int l15     = lane & 15;

    // This lane's A-matrix row = edge (ebase + l15)'s source node features.
    const long long s = src[ebase + l15];
    const float* __restrict__ hrow = h + s * HID;

    // Contiguous {k, k+1} pairs of this lane's B column, as v2f for ds_load_b64.
    const v2f* __restrict__ w2row = (const v2f*)&w2s[l15][0];

    v8f acc = {0.f, 0.f, 0.f, 0.f, 0.f, 0.f, 0.f, 0.f};
#pragma unroll
    for (int kk = 0; kk < 32; ++kk) {
        const int k = kk * 4 + halfsel * 2;
        v2f a = *(const v2f*)(hrow + k);     // 8B-aligned (k even)
        v2f b = w2row[kk * 2 + halfsel];     // elements {k, k+1}
        acc = __builtin_amdgcn_wmma_f32_16x16x4_f32(
            false, a, false, b, (short)0, acc, false, false);
    }

    // Scatter: lane column l15 < 10 holds class l15 for 8 edges.
    if (l15 < N_CLS) {
#pragma unroll
        for (int r = 0; r < 8; ++r) {
            const long long e = ebase + r + halfsel * 8;
            const long long d = dst[e];
            __hip_atomic_fetch_add(&out[d * N_CLS + l15], acc[r],
                                   __ATOMIC_RELAXED, __HIP_MEMORY_SCOPE_AGENT);
        }
    }
}

// ---------------------------------------------------------------------------
extern "C" void kernel_launch(void* const* d_in, const int* in_sizes, int n_in,
                              void* d_out, int out_size, void* d_ws, size_t ws_size,
                              hipStream_t stream) {
    const float*     x    = (const float*)d_in[0];
    const long long* eidx = (const long long*)d_in[1];   // int64 [2, N_EDGES]
    const float*     W1   = (const float*)d_in[2];
    const float*     b1   = (const float*)d_in[3];
    const float*     W2   = (const float*)d_in[4];
    const float*     b2   = (const float*)d_in[5];
    float*           out  = (float*)d_out;

    const long long* src  = eidx;
    const long long* dstp = eidx + N_EDGES;

    float* h = (float*)d_ws;                              // 200000*128*4 = 102.4 MB

    // out = b2 (broadcast)
    {
        const int n = N_NODES * N_CLS;
        gcn_init_out<<<(n + 255) / 256, 256, 0, stream>>>(b2, out, n);
    }
    // h = relu(x @ W1 + b1): 12500 waves, 4 waves/block -> 3125 blocks
    gcn_lin1_relu<<<N_NODES / 16 / 4, 128, 0, stream>>>(x, W1, b1, h);
    // edge phase: 75000 waves, 8 waves/block -> 9375 blocks
    gcn_edge_scatter<<<N_EDGES / 16 / 8, 256, 0, stream>>>(h, src, dstp, W2, out);
}